// Encoder_25692494364677
// MI455X (gfx1250) — compile-verified
//
#include <hip/hip_runtime.h>
#include <hip/hip_bf16.h>
#include <stdint.h>

typedef __attribute__((ext_vector_type(2))) float v2f;
typedef __attribute__((ext_vector_type(8))) float v8f;

#define N_NODES   8192
#define N_EDGES   65536
#define N_TRIPLES (N_NODES + N_EDGES)   /* 73728, divisible by 16 */
#define DIM       768
#define K3        (3 * DIM)             /* 2304 */
#define NREL      38
#define BATCH     8
#define SLEN      64
#define KC        64                    /* K-chunk staged in LDS */

/* ---------------- kernel 1: padded relation table (R+1, 768) ---------------- */
__global__ void build_rel_ext(const float* __restrict__ rel_emb,
                              const float* __restrict__ self_loop,
                              float* __restrict__ rel_ext) {
    int i = blockIdx.x * blockDim.x + threadIdx.x;
    const int total = (NREL + 1) * DIM;
    if (i >= total) return;
    int row = i / DIM;
    int col = i - row * DIM;
    rel_ext[i] = (row < NREL) ? rel_emb[i] : self_loop[col];
}

/* ------- kernel 2: per-triple metadata + triple_ids output (as float) ------- */
__global__ void build_triples(const int* __restrict__ concept_ids,
                              const int* __restrict__ edge_index,
                              const float* __restrict__ edge_attr,
                              int* __restrict__ hcid, int* __restrict__ tcid,
                              int* __restrict__ rrel, float* __restrict__ wext,
                              float* __restrict__ out_ids) {
    int t = blockIdx.x * blockDim.x + threadIdx.x;
    if (t >= N_TRIPLES) return;
    int head, tail, rel;
    float w;
    if (t < N_EDGES) {
        head = edge_index[t];
        tail = edge_index[N_EDGES + t];
        rel  = (int)edge_attr[2 * t];
        w    = edge_attr[2 * t + 1];
    } else {
        head = tail = t - N_EDGES;
        rel  = NREL;          /* self-loop id */
        w    = 1.0f;
    }
    int hc = concept_ids[head];
    int tc = concept_ids[tail];
    hcid[t] = hc; tcid[t] = tc; rrel[t] = rel; wext[t] = w;
    out_ids[3 * t + 0] = (float)hc;
    out_ids[3 * t + 1] = (float)rel;
    out_ids[3 * t + 2] = (float)tc;
}

/* ---------------- kernel 3: (B, T) membership mask ---------------- */
__global__ void build_mask(const int* __restrict__ sent_ids,
                           const int* __restrict__ hcid,
                           const int* __restrict__ tcid,
                           float* __restrict__ out_mask) {
    __shared__ int sent[BATCH * SLEN];
    for (int i = threadIdx.x; i < BATCH * SLEN; i += blockDim.x)
        sent[i] = sent_ids[i];
    __syncthreads();
    int t = blockIdx.x * blockDim.x + threadIdx.x;
    if (t >= N_TRIPLES) return;
    int hc = hcid[t];
    int tc = tcid[t];
    #pragma unroll
    for (int b = 0; b < BATCH; ++b) {
        int any = 0;
        #pragma unroll 8
        for (int s = 0; s < SLEN; ++s) {
            int v = sent[b * SLEN + s];
            any |= (v == hc) | (v == tc);
        }
        out_mask[(size_t)b * N_TRIPLES + t] = any ? 1.0f : 0.0f;
    }
}

/* ---------------- kernel 4: WMMA GEMM (T x 2304) @ (2304 x 768) + bias ------
 * grid = T/16 workgroups, 192 threads (6 waves).  Each workgroup: 16-row tile.
 * Each wave: 128-wide N slice = 8 accumulator tiles of 16x16 f32.
 * A tile assembled per 64-wide K-chunk in LDS from gathered embeddings.    */
__global__ __launch_bounds__(192)
void gemm_wmma(const float* __restrict__ cemb,
               const float* __restrict__ W,
               const float* __restrict__ bias,
               const float* __restrict__ rel_ext,
               const int* __restrict__ hcid, const int* __restrict__ tcid,
               const int* __restrict__ rrel, const float* __restrict__ wext,
               float* __restrict__ out) {
    __shared__ float As[16][KC + 4];     /* stride 68 dwords: conflict-free */
    __shared__ int   rowH[16], rowT[16], rowR[16];
    __shared__ float rowW[16];

    const int tid  = threadIdx.x;
    const int lane = tid & 31;
    const int wave = tid >> 5;           /* 0..5 */
    const int ln   = lane & 15;
    const int hi   = lane >> 4;          /* 0 or 1 */
    const int t0   = blockIdx.x * 16;
    const int n_base = wave * 128;

    if (tid < 16) {
        int t = t0 + tid;
        rowH[tid] = hcid[t];
        rowT[tid] = tcid[t];
        rowR[tid] = rrel[t];
        rowW[tid] = wext[t];
    }
    __syncthreads();

    /* accumulators, bias-initialized */
    v8f acc[8];
    #pragma unroll
    for (int nt = 0; nt < 8; ++nt) {
        float bv = bias[n_base + nt * 16 + ln];
        acc[nt] = (v8f){bv, bv, bv, bv, bv, bv, bv, bv};
    }

    for (int kc = 0; kc < K3; kc += KC) {
        /* ---- stage A chunk: 16 rows x 64 K into LDS (gather + scale) ---- */
        const int seg = kc / DIM;              /* 0: head, 1: relation, 2: tail */
        const int kin = kc - seg * DIM;
        for (int idx = tid; idx < 16 * KC; idx += 192) {
            int r = idx >> 6;
            int c = idx & (KC - 1);
            float v;
            if (seg == 0)      v = cemb[(size_t)rowH[r] * DIM + kin + c];
            else if (seg == 2) v = cemb[(size_t)rowT[r] * DIM + kin + c];
            else               v = rowW[r] * rel_ext[rowR[r] * DIM + kin + c];
            As[r][c] = v;
        }
        /* hint-prefetch next W K-chunk into cache (global_prefetch_b8) */
        if (kc + KC < K3)
            __builtin_prefetch(W + (size_t)(kc + KC) * DIM + (size_t)tid * 256, 0, 1);
        __syncthreads();

        /* ---- compute: 16 k-steps x 8 N tiles of v_wmma_f32_16x16x4_f32 ---- */
        #pragma unroll 4
        for (int k = 0; k < KC; k += 4) {
            v2f a = *(const v2f*)&As[ln][k + 2 * hi];        /* M=ln, K=2*hi+{0,1} */
            const int kk = kc + k + 2 * hi;
            const float* w0 = W + (size_t)kk * DIM + n_base + ln;
            #pragma unroll
            for (int nt = 0; nt < 8; ++nt) {
                v2f bb;
                bb.x = w0[nt * 16];          /* W[kk,   n] */
                bb.y = w0[DIM + nt * 16];    /* W[kk+1, n] */
                acc[nt] = __builtin_amdgcn_wmma_f32_16x16x4_f32(
                    false, a, false, bb, (short)0, acc[nt], false, false);
            }
        }
        __syncthreads();
    }

    /* ---- store: lane L, vgpr r -> (m = r + 8*hi, n = n_base + nt*16 + ln) ---- */
    #pragma unroll
    for (int nt = 0; nt < 8; ++nt) {
        const int n = n_base + nt * 16 + ln;
        #pragma unroll
        for (int r = 0; r < 8; ++r) {
            out[(size_t)(t0 + r + 8 * hi) * DIM + n] = acc[nt][r];
        }
    }
}

/* ------------------------------- launcher ------------------------------- */
extern "C" void kernel_launch(void* const* d_in, const int* in_sizes, int n_in,
                              void* d_out, int out_size, void* d_ws, size_t ws_size,
                              hipStream_t stream) {
    const int*   concept_ids = (const int*)d_in[0];
    const int*   edge_index  = (const int*)d_in[1];
    const float* edge_attr   = (const float*)d_in[2];
    const int*   sent_ids    = (const int*)d_in[3];
    const float* cemb        = (const float*)d_in[4];
    const float* remb        = (const float*)d_in[5];
    const float* sloop       = (const float*)d_in[6];
    const float* W           = (const float*)d_in[7];
    const float* bias        = (const float*)d_in[8];

    float* out_enc  = (float*)d_out;
    float* out_mask = out_enc + (size_t)N_TRIPLES * DIM;
    float* out_ids  = out_mask + (size_t)BATCH * N_TRIPLES;

    char* ws = (char*)d_ws;
    float* rel_ext = (float*)ws;  ws += (size_t)(NREL + 1) * DIM * sizeof(float);
    int*   hcid    = (int*)ws;    ws += (size_t)N_TRIPLES * sizeof(int);
    int*   tcid    = (int*)ws;    ws += (size_t)N_TRIPLES * sizeof(int);
    int*   rrel    = (int*)ws;    ws += (size_t)N_TRIPLES * sizeof(int);
    float* wext    = (float*)ws;  ws += (size_t)N_TRIPLES * sizeof(float);

    {   /* padded relation table */
        int total = (NREL + 1) * DIM;
        build_rel_ext<<<(total + 255) / 256, 256, 0, stream>>>(remb, sloop, rel_ext);
    }
    build_triples<<<(N_TRIPLES + 255) / 256, 256, 0, stream>>>(
        concept_ids, edge_index, edge_attr, hcid, tcid, rrel, wext, out_ids);
    build_mask<<<(N_TRIPLES + 255) / 256, 256, 0, stream>>>(
        sent_ids, hcid, tcid, out_mask);
    gemm_wmma<<<N_TRIPLES / 16, 192, 0, stream>>>(
        cemb, W, bias, rel_ext, hcid, tcid, rrel, wext, out_enc);
}